// HeteroGNN_66846870994984
// MI455X (gfx1250) — compile-verified
//
#include <hip/hip_runtime.h>
#include <hip/hip_bf16.h>
#include <stdint.h>

// ---------------------------------------------------------------------------
// HeteroGNN (2x bipartite GATConv host->flow, summed) + ReLU + Linear(256->8)
// MI455X / gfx1250, wave32. fp32 WMMA (V_WMMA_F32_16X16X4_F32) for the GEMM.
// ---------------------------------------------------------------------------

typedef float v2f __attribute__((ext_vector_type(2)));
typedef float v8f __attribute__((ext_vector_type(8)));

#define NEG_SLOPE 0.2f

// Order-preserving float -> uint map for atomicMax-based segment max.
__device__ __forceinline__ unsigned enc_f(float f) {
    unsigned u = __float_as_uint(f);
    return (u & 0x80000000u) ? ~u : (u | 0x80000000u);
}
__device__ __forceinline__ float dec_f(unsigned e) {
    return (e & 0x80000000u) ? __uint_as_float(e ^ 0x80000000u)
                             : __uint_as_float(~e);
}

// -------------------- init kernels --------------------
__global__ void k_init_flow(float* __restrict__ flow, const float* __restrict__ bias,
                            int Nf, int H) {
    size_t i = (size_t)blockIdx.x * blockDim.x + threadIdx.x;
    size_t n = (size_t)Nf * H;
    if (i >= n) return;
    int h = (int)(i % H);
    flow[i] = bias[h] + bias[H + h];  // bias of both relations (HeteroConv sum)
}

__global__ void k_init_seg(unsigned* __restrict__ menc, float* __restrict__ den, int n) {
    int i = blockIdx.x * blockDim.x + threadIdx.x;
    if (i >= n) return;
    menc[i] = 0x007FFFFFu;  // enc(-inf)
    den[i]  = 0.0f;
}

// -------------------- attention-vector folding --------------------
// wsvec[r][k] = sum_h Wsrc[r][k][h]*att_src[r][h]   (k < Dh)
// wdvec[r][d] = sum_h Wdst[r][d][h]*att_dst[r][h]   (d < Df)
__global__ void k_prep(const float* __restrict__ Wsrc, const float* __restrict__ Wdst,
                       const float* __restrict__ as, const float* __restrict__ ad,
                       float* __restrict__ wsvec, float* __restrict__ wdvec,
                       int Dh, int Df, int H) {
    int t = blockIdx.x * blockDim.x + threadIdx.x;
    int nws = 2 * Dh;
    if (t < nws) {
        int r = t / Dh, k = t - r * Dh;
        const float* wrow = Wsrc + ((size_t)r * Dh + k) * H;
        const float* av   = as + (size_t)r * H;
        float s = 0.f;
        for (int h = 0; h < H; ++h) s += wrow[h] * av[h];
        wsvec[t] = s;
    } else if (t < nws + 2 * Df) {
        int t2 = t - nws;
        int r = t2 / Df, d = t2 - r * Df;
        const float* wrow = Wdst + ((size_t)r * Df + d) * H;
        const float* av   = ad + (size_t)r * H;
        float s = 0.f;
        for (int h = 0; h < H; ++h) s += wrow[h] * av[h];
        wdvec[t2] = s;
    }
}

// -------------------- hs = x_host @ Wsrc[r], fp32 WMMA --------------------
// One wave computes a 16(M) x 64(N) strip: 4 accumulators sharing one A frag,
// K loop in steps of 4 using v_wmma_f32_16x16x4_f32.
// A(16x4): lanes 0-15 -> M rows, VGPR0=K(2*hi), VGPR1=K(2*hi+1); contiguous ->
//   one float2 load. B(4x16): mirrored. C/D: VGPR v -> M=v / M=v+8.
// Out-of-range M rows are CLAMPED (not predicated): they only pollute output
// rows that the guarded store never writes, and clamping keeps all loads
// unconditional (no exec-mask save/restore in the inner loop).
__global__ void k_gemm_hs(const float* __restrict__ A, const float* __restrict__ W,
                          float* __restrict__ hs, int Nh, int Dh, int H,
                          int mt, int ng) {
    int wave = (blockIdx.x * blockDim.x + threadIdx.x) >> 5;
    int lane = threadIdx.x & 31;
    int strips = 2 * mt * ng;          // ng = number of 64-wide N groups
    if (wave >= strips) return;
    int r   = wave / (mt * ng);
    int rem = wave - r * (mt * ng);
    int mi = rem / ng, gi = rem - mi * ng;
    int m0 = mi * 16, n0 = gi * 64;
    int jmax = (H - n0) / 16; if (jmax > 4) jmax = 4;

    int mrow = m0 + (lane & 15);
    if (mrow >= Nh) mrow = Nh - 1;     // clamp (see note above)
    int nc = n0 + (lane & 15);
    int hi = lane >> 4;                // 0 or 1

    const float* ap = A + (size_t)mrow * Dh + 2 * hi;           // += 4 per step
    const float* bp = W + (size_t)r * Dh * H + (size_t)(2 * hi) * H + nc; // += 4H

    v8f c0 = {}, c1 = {}, c2 = {}, c3 = {};
    for (int k0 = 0; k0 < Dh; k0 += 4) {
        v2f a = *(const v2f*)ap;       // K = 2*hi, 2*hi+1 (contiguous)
        v2f b0, b1, b2, b3;
        b0.x = bp[0];   b0.y = bp[H];
        b1.x = bp[16];  b1.y = bp[H + 16];
        b2.x = bp[32];  b2.y = bp[H + 32];
        b3.x = bp[48];  b3.y = bp[H + 48];
        c0 = __builtin_amdgcn_wmma_f32_16x16x4_f32(false, a, false, b0, (short)0, c0, false, false);
        c1 = __builtin_amdgcn_wmma_f32_16x16x4_f32(false, a, false, b1, (short)0, c1, false, false);
        c2 = __builtin_amdgcn_wmma_f32_16x16x4_f32(false, a, false, b2, (short)0, c2, false, false);
        c3 = __builtin_amdgcn_wmma_f32_16x16x4_f32(false, a, false, b3, (short)0, c3, false, false);
        ap += 4;
        bp += (size_t)4 * H;
    }

    float* hsr = hs + (size_t)r * Nh * H;
#pragma unroll
    for (int v = 0; v < 8; ++v) {
        int row = m0 + v + 8 * hi;
        if (row >= Nh) continue;
        float* orow = hsr + (size_t)row * H + nc;
        orow[0] = c0[v];
        if (jmax > 1) orow[16] = c1[v];
        if (jmax > 2) orow[32] = c2[v];
        if (jmax > 3) orow[48] = c3[v];
    }
}

// -------------------- alpha_src[r][n] = dot(x_host[n], wsvec[r]) --------------------
__global__ void k_alpha_src(const float* __restrict__ x, const float* __restrict__ wsvec,
                            float* __restrict__ asrc, int Nh, int Dh) {
    int wave = (blockIdx.x * blockDim.x + threadIdx.x) >> 5;
    int lane = threadIdx.x & 31;
    if (wave >= Nh) return;
    const float* row = x + (size_t)wave * Dh;
    float s0 = 0.f, s1 = 0.f;
    for (int h = lane * 2; h < Dh; h += 64) {
        float2 v  = *(const float2*)(row + h);
        float2 w0 = *(const float2*)(wsvec + h);
        float2 w1 = *(const float2*)(wsvec + Dh + h);
        s0 += v.x * w0.x + v.y * w0.y;
        s1 += v.x * w1.x + v.y * w1.y;
    }
    for (int off = 16; off; off >>= 1) {
        s0 += __shfl_xor(s0, off);
        s1 += __shfl_xor(s1, off);
    }
    if (lane == 0) { asrc[wave] = s0; asrc[(size_t)Nh + wave] = s1; }
}

// -------------------- alpha_dst[r][f] = dot(x_flow[f], wdvec[r]) --------------------
__global__ void k_alpha_dst(const float* __restrict__ x, const float* __restrict__ wdvec,
                            float* __restrict__ adst, int Nf, int Df) {
    int wave = (blockIdx.x * blockDim.x + threadIdx.x) >> 5;
    int lane = threadIdx.x & 31;
    if (wave >= Nf) return;
    const float* row = x + (size_t)wave * Df;
    float s0 = 0.f, s1 = 0.f;
    for (int h = lane * 4; h < Df; h += 128) {
        float4 v  = *(const float4*)(row + h);
        float4 w0 = *(const float4*)(wdvec + h);
        float4 w1 = *(const float4*)(wdvec + Df + h);
        s0 += v.x * w0.x + v.y * w0.y + v.z * w0.z + v.w * w0.w;
        s1 += v.x * w1.x + v.y * w1.y + v.z * w1.z + v.w * w1.w;
    }
    for (int off = 16; off; off >>= 1) {
        s0 += __shfl_xor(s0, off);
        s1 += __shfl_xor(s1, off);
    }
    if (lane == 0) { adst[wave] = s0; adst[(size_t)Nf + wave] = s1; }
}

// -------------------- edge pass 1: leaky-relu logit + segment max --------------------
__global__ void k_edge1(const float* __restrict__ asrc, const float* __restrict__ adst,
                        const int* __restrict__ src, const int* __restrict__ dst,
                        float* __restrict__ ebuf, unsigned* __restrict__ menc,
                        int Nh, int Nf, int E) {
    int t = blockIdx.x * blockDim.x + threadIdx.x;
    if (t >= 2 * E) return;
    int r = t / E, e = t - r * E;
    int s = src[(size_t)r * E + e];
    int d = dst[(size_t)r * E + e];
    float v = asrc[(size_t)r * Nh + s] + adst[(size_t)r * Nf + d];
    v = (v > 0.f) ? v : NEG_SLOPE * v;
    ebuf[t] = v;
    atomicMax(menc + (size_t)r * Nf + d, enc_f(v));
}

// -------------------- edge pass 2: exp + segment sum --------------------
__global__ void k_edge2(float* __restrict__ ebuf, const unsigned* __restrict__ menc,
                        const int* __restrict__ dst, float* __restrict__ den,
                        int Nf, int E) {
    int t = blockIdx.x * blockDim.x + threadIdx.x;
    if (t >= 2 * E) return;
    int r = t / E, e = t - r * E;
    int d = dst[(size_t)r * E + e];
    float m = dec_f(menc[(size_t)r * Nf + d]);
    float x = __expf(ebuf[t] - m);
    ebuf[t] = x;  // in-place: ebuf now holds exp(e - m)
    atomicAdd(den + (size_t)r * Nf + d, x);
}

// -------------------- edge pass 3: weighted scatter-add (wave per edge) --------------
__global__ void k_agg(const float* __restrict__ hs, const float* __restrict__ ex,
                      const float* __restrict__ den, const int* __restrict__ src,
                      const int* __restrict__ dst, float* __restrict__ flow,
                      int Nh, int Nf, int H, int E) {
    int wave = (blockIdx.x * blockDim.x + threadIdx.x) >> 5;
    int lane = threadIdx.x & 31;
    if (wave >= 2 * E) return;
    int r = wave / E, e = wave - r * E;
    int s = src[(size_t)r * E + e];
    int d = dst[(size_t)r * E + e];
    float dn = den[(size_t)r * Nf + d];            // > 0: this edge contributed
    float alpha = ex[wave] / dn;
    const float4* hrow = (const float4*)(hs + ((size_t)r * Nh + s) * H);
    float* frow = flow + (size_t)d * H;
    for (int h4 = lane; h4 * 4 < H; h4 += 32) {
        float4 v = hrow[h4];
        int h = h4 * 4;
        atomicAdd(frow + h + 0, alpha * v.x);
        atomicAdd(frow + h + 1, alpha * v.y);
        atomicAdd(frow + h + 2, alpha * v.z);
        atomicAdd(frow + h + 3, alpha * v.w);
    }
}

// -------------------- out = relu(flow) @ linW + linb (wave per node) --------------
__global__ void k_out(const float* __restrict__ flow, const float* __restrict__ linW,
                      const float* __restrict__ linb, float* __restrict__ out,
                      int Nf, int H, int C) {
    int wave = (blockIdx.x * blockDim.x + threadIdx.x) >> 5;
    int lane = threadIdx.x & 31;
    if (wave >= Nf) return;
    const float* row = flow + (size_t)wave * H;
    float acc[8];
#pragma unroll
    for (int c = 0; c < 8; ++c) acc[c] = 0.f;
    for (int h = lane; h < H; h += 32) {
        float v = row[h];
        v = (v > 0.f) ? v : 0.f;  // ReLU
        const float* wr = linW + (size_t)h * C;
        for (int c = 0; c < C; ++c) acc[c] += v * wr[c];
    }
    for (int off = 16; off; off >>= 1)
        for (int c = 0; c < C; ++c) acc[c] += __shfl_xor(acc[c], off);
    if (lane == 0)
        for (int c = 0; c < C; ++c) out[(size_t)wave * C + c] = acc[c] + linb[c];
}

// ---------------------------------------------------------------------------
extern "C" void kernel_launch(void* const* d_in, const int* in_sizes, int n_in,
                              void* d_out, int out_size, void* d_ws, size_t ws_size,
                              hipStream_t stream) {
    const float* x_host  = (const float*)d_in[0];
    const float* x_flow  = (const float*)d_in[1];
    const int*   srcH    = (const int*)d_in[2];
    const int*   dstF    = (const int*)d_in[3];
    const float* Wsrc    = (const float*)d_in[4];
    const float* Wdst    = (const float*)d_in[5];
    const float* att_src = (const float*)d_in[6];
    const float* att_dst = (const float*)d_in[7];
    const float* bias    = (const float*)d_in[8];
    const float* linW    = (const float*)d_in[9];
    const float* linb    = (const float*)d_in[10];
    float* out = (float*)d_out;

    const int H  = in_sizes[6] / 2;           // att_src: (2,H)
    const int C  = in_sizes[10];              // linb: (C,)
    const int Dh = in_sizes[4] / (2 * H);     // Wsrc: (2,Dh,H)
    const int Df = in_sizes[5] / (2 * H);     // Wdst: (2,Df,H)
    const int Nh = in_sizes[0] / Dh;
    const int Nf = in_sizes[1] / Df;
    const int E  = in_sizes[2] / 2;

    // workspace carve-out (256B aligned)
    char* p = (char*)d_ws;
    auto carve = [&](size_t bytes) -> void* {
        void* r = (void*)p;
        p += (bytes + 255) & ~(size_t)255;
        return r;
    };
    float*    hs    = (float*)   carve((size_t)2 * Nh * H * sizeof(float));
    float*    flow  = (float*)   carve((size_t)Nf * H * sizeof(float));
    float*    asrc  = (float*)   carve((size_t)2 * Nh * sizeof(float));
    float*    adst  = (float*)   carve((size_t)2 * Nf * sizeof(float));
    unsigned* menc  = (unsigned*)carve((size_t)2 * Nf * sizeof(unsigned));
    float*    den   = (float*)   carve((size_t)2 * Nf * sizeof(float));
    float*    ebuf  = (float*)   carve((size_t)2 * E * sizeof(float));
    float*    wsvec = (float*)   carve((size_t)2 * Dh * sizeof(float));
    float*    wdvec = (float*)   carve((size_t)2 * Df * sizeof(float));

    const int BS = 256;

    // 1. init accumulators
    {
        size_t n = (size_t)Nf * H;
        k_init_flow<<<(unsigned)((n + BS - 1) / BS), BS, 0, stream>>>(flow, bias, Nf, H);
        int n2 = 2 * Nf;
        k_init_seg<<<(n2 + BS - 1) / BS, BS, 0, stream>>>(menc, den, n2);
    }
    // 2. fold attention vectors
    {
        int n = 2 * Dh + 2 * Df;
        k_prep<<<(n + BS - 1) / BS, BS, 0, stream>>>(Wsrc, Wdst, att_src, att_dst,
                                                     wsvec, wdvec, Dh, Df, H);
    }
    // 3. hs = x_host @ Wsrc[r]  (fp32 WMMA, 16x64 strip per wave)
    {
        int mt = (Nh + 15) / 16;
        int ng = (H + 63) / 64;
        int waves = 2 * mt * ng;
        k_gemm_hs<<<(waves * 32 + BS - 1) / BS, BS, 0, stream>>>(x_host, Wsrc, hs,
                                                                 Nh, Dh, H, mt, ng);
    }
    // 4. per-node attention scalars
    k_alpha_src<<<((size_t)Nh * 32 + BS - 1) / BS, BS, 0, stream>>>(x_host, wsvec, asrc, Nh, Dh);
    k_alpha_dst<<<((size_t)Nf * 32 + BS - 1) / BS, BS, 0, stream>>>(x_flow, wdvec, adst, Nf, Df);
    // 5-6. segment softmax over edges
    {
        int n = 2 * E;
        k_edge1<<<(n + BS - 1) / BS, BS, 0, stream>>>(asrc, adst, srcH, dstF, ebuf, menc, Nh, Nf, E);
        k_edge2<<<(n + BS - 1) / BS, BS, 0, stream>>>(ebuf, menc, dstF, den, Nf, E);
    }
    // 7. weighted message scatter-add
    {
        size_t threads = (size_t)2 * E * 32;
        k_agg<<<(unsigned)((threads + BS - 1) / BS), BS, 0, stream>>>(hs, ebuf, den, srcH, dstF,
                                                                     flow, Nh, Nf, H, E);
    }
    // 8. ReLU + linear head
    {
        size_t threads = (size_t)Nf * 32;
        k_out<<<(unsigned)((threads + BS - 1) / BS), BS, 0, stream>>>(flow, linW, linb, out, Nf, H, C);
    }
}